// OnlineFlashMTPModel_49022756716957
// MI455X (gfx1250) — compile-verified
//
#include <hip/hip_runtime.h>
#include <hip/hip_bf16.h>
#include <stdint.h>

typedef __attribute__((ext_vector_type(2))) float v2f;
typedef __attribute__((ext_vector_type(8))) float v8f;
typedef int v4i_ __attribute__((vector_size(4 * sizeof(int))));

#define Bb 2
#define Ss 4096
#define Hh 2048
#define Ll 3
#define Nn 512
#define Vv 32000
#define BS 16
#define HD 128
#define NA (Bb*Nn)          // 1024 anchors
#define MASK_ID (Vv-1)

// ---- CDNA5 async global->LDS copy (guarded; fallback = sync reg roundtrip) ----
#if defined(__gfx1250__) && __has_builtin(__builtin_amdgcn_global_load_async_to_lds_b128) && __has_builtin(__builtin_amdgcn_s_wait_asynccnt)
#define USE_ASYNC 1
#define AS1 __attribute__((address_space(1)))
#define AS3 __attribute__((address_space(3)))
__device__ __forceinline__ void async_cp16(const float* g, float* l) {
    __builtin_amdgcn_global_load_async_to_lds_b128((AS1 v4i_*)g, (AS3 v4i_*)l, 0, 0);
}
#else
#define USE_ASYNC 0
__device__ __forceinline__ void async_cp16(const float* g, float* l) {
    const float4 v = *(const float4*)g;
    l[0] = v.x; l[1] = v.y; l[2] = v.z; l[3] = v.w;
}
#endif

// ---------------- Phase 0: build unique-row matrix U (NA*5, H) ----------------
__global__ __launch_bounds__(256)
void build_u_kernel(const int* __restrict__ input_ids,
                    const float* __restrict__ hidden,
                    const int* __restrict__ anchors,
                    const unsigned char* __restrict__ keep,
                    const float* __restrict__ embed,
                    float* __restrict__ U)
{
    int a = blockIdx.x;
    int b = a >> 9;
    int anc = anchors[a];
    int ctx = anc - 1; if (ctx < 0) ctx = 0;
    int ancc = anc; if (ancc < 0) ancc = 0; if (ancc > Ss-1) ancc = Ss-1;
    int tok = input_ids[b*Ss + ancc];
    int first = keep[a] ? tok : MASK_ID;

    float* ub = U + (size_t)a * 5 * Hh;
    for (int h = threadIdx.x; h < Hh; h += blockDim.x) {
        #pragma unroll
        for (int l = 0; l < Ll; ++l)
            ub[(size_t)l*Hh + h] = hidden[(((size_t)l*Bb + b)*Ss + ctx)*Hh + h];
        ub[(size_t)3*Hh + h] = embed[(size_t)first*Hh + h];
        ub[(size_t)4*Hh + h] = embed[(size_t)MASK_ID*Hh + h];
    }
}

// ---------------- WMMA fp32 GEMM: Y(M,Nt) = A(M,K) * W(K,Nt) ----------------
// 256 threads = 8 waves, WG tile 64x128, K-chunk 32, double-buffered async LDS.
// A: [64][36] pad -> conflict-free b64 reads, 16B-aligned b128 stages.
// B: [32][136] pad -> half-wave K-phases hit disjoint bank groups.
#define TM 64
#define TN 128
#define TK 32
#define ASTR 36
#define BSTR 136

__global__ __launch_bounds__(256)
void gemm_f32_wmma_kernel(const float* __restrict__ A,
                          const float* __restrict__ W,
                          float* __restrict__ Y,
                          int M, int Nt, int K)
{
    __shared__ __align__(16) float As[2][TM*ASTR];
    __shared__ __align__(16) float Bs[2][TK*BSTR];

    const int tid  = threadIdx.x;
    const int wave = tid >> 5;
    const int lane = tid & 31;
    const int m0 = blockIdx.y * TM;
    const int nblk = blockIdx.x * TN;

    const int mrow   = lane & 15;
    const int kphase = (lane >> 4) * 2;         // 0 or 2
    const int bcol   = wave*16 + (lane & 15);

    // staging coordinates (per thread): 2 A-chunks + 4 B-chunks of 16B
    const int ar0 = tid >> 3,        ac0 = (tid & 7) << 2;        // A slots 0..255
    const int ar1 = (tid+256) >> 3,  ac1 = ac0;                   // A slots 256..511
    const int br  = tid >> 5,        bc  = (tid & 31) << 2;       // B row 0..7, col

    v8f acc[4];
    #pragma unroll
    for (int i = 0; i < 4; ++i) acc[i] = (v8f){0.f,0.f,0.f,0.f,0.f,0.f,0.f,0.f};

    const float* Abase = A + (size_t)m0 * K;
    const float* Bbase = W + nblk;

    // issue stage for chunk 0 into buffer 0
    {
        const float* Ag = Abase;
        const float* Bg = Bbase;
        async_cp16(Ag + (size_t)ar0*K + ac0, &As[0][ar0*ASTR + ac0]);
        async_cp16(Ag + (size_t)ar1*K + ac1, &As[0][ar1*ASTR + ac1]);
        #pragma unroll
        for (int i = 0; i < 4; ++i)
            async_cp16(Bg + (size_t)(br + i*8)*Nt + bc, &Bs[0][(br + i*8)*BSTR + bc]);
    }

    int buf = 0;
    for (int k0 = 0; k0 < K; k0 += TK) {
        const bool more = (k0 + TK) < K;
        if (more) {                     // prefetch next chunk into other buffer
            const float* Ag = Abase + (k0 + TK);
            const float* Bg = Bbase + (size_t)(k0 + TK) * Nt;
            async_cp16(Ag + (size_t)ar0*K + ac0, &As[buf^1][ar0*ASTR + ac0]);
            async_cp16(Ag + (size_t)ar1*K + ac1, &As[buf^1][ar1*ASTR + ac1]);
            #pragma unroll
            for (int i = 0; i < 4; ++i)
                async_cp16(Bg + (size_t)(br + i*8)*Nt + bc, &Bs[buf^1][(br + i*8)*BSTR + bc]);
        }
#if USE_ASYNC
        if (more) __builtin_amdgcn_s_wait_asynccnt(6);   // current chunk landed
        else      __builtin_amdgcn_s_wait_asynccnt(0);
#endif
        __syncthreads();

        const float* as = As[buf];
        const float* bs = Bs[buf];
        #pragma unroll
        for (int kk = 0; kk < TK; kk += 4) {
            v2f bfrag;
            bfrag.x = bs[(kk + kphase    )*BSTR + bcol];
            bfrag.y = bs[(kk + kphase + 1)*BSTR + bcol];
            #pragma unroll
            for (int ms = 0; ms < 4; ++ms) {
                v2f afrag;
                afrag.x = as[(ms*16 + mrow)*ASTR + kk + kphase];
                afrag.y = as[(ms*16 + mrow)*ASTR + kk + kphase + 1];
                acc[ms] = __builtin_amdgcn_wmma_f32_16x16x4_f32(
                    false, afrag, false, bfrag, (short)0, acc[ms], false, false);
            }
        }
        __syncthreads();               // all reads done before buffer is restaged
        buf ^= 1;
    }

    // D layout: n = lane&15, m = vgpr + 8*(lane>>4)
    const int nc  = nblk + bcol;
    const int mhi = (lane >> 4) * 8;
    #pragma unroll
    for (int ms = 0; ms < 4; ++ms) {
        #pragma unroll
        for (int r = 0; r < 8; ++r) {
            int mr = m0 + ms*16 + mhi + r;
            Y[(size_t)mr * Nt + nc] = acc[ms][r];
        }
    }
}

// ---------------- Phase 2: per-anchor attention with key multiplicity ----------------
__device__ inline float wave_sum32(float v) {
    #pragma unroll
    for (int off = 16; off > 0; off >>= 1) v += __shfl_xor(v, off, 32);
    return v;
}

__global__ __launch_bounds__(256)
void attn_kernel(const float* __restrict__ Qm, const float* __restrict__ Km,
                 const float* __restrict__ Vm, const unsigned char* __restrict__ keep,
                 float* __restrict__ AU)
{
    const int a    = blockIdx.x;
    const int wave = threadIdx.x >> 5;
    const int lane = threadIdx.x & 31;
    const bool kp  = keep[a] != 0;
    const float scale = 0.08838834764831845f;   // 1/sqrt(128)
    const float wmul[5] = {1.f, 1.f, 1.f, 1.f, 15.f};

    const size_t rowbase = (size_t)a * 5 * Hh;

    #pragma unroll
    for (int hh = 0; hh < 2; ++hh) {
        const int h = wave*2 + hh;
        const int col0 = h*HD + lane*4;

        float q[2][4], k[5][4], v[5][4];
        #pragma unroll
        for (int d = 0; d < 4; ++d) {
            q[0][d] = Qm[rowbase + (size_t)3*Hh + col0 + d];
            q[1][d] = Qm[rowbase + (size_t)4*Hh + col0 + d];
        }
        #pragma unroll
        for (int j = 0; j < 5; ++j)
            #pragma unroll
            for (int d = 0; d < 4; ++d) {
                k[j][d] = Km[rowbase + (size_t)j*Hh + col0 + d];
                v[j][d] = Vm[rowbase + (size_t)j*Hh + col0 + d];
            }

        float s[2][5];
        #pragma unroll
        for (int qi = 0; qi < 2; ++qi)
            #pragma unroll
            for (int j = 0; j < 5; ++j) {
                float p = 0.f;
                #pragma unroll
                for (int d = 0; d < 4; ++d) p += q[qi][d]*k[j][d];
                s[qi][j] = wave_sum32(p) * scale;
            }

        #pragma unroll
        for (int qi = 0; qi < 2; ++qi) {
            float mx = s[qi][0];
            #pragma unroll
            for (int j = 1; j < 5; ++j) mx = fmaxf(mx, s[qi][j]);
            float e[5]; float den = 0.f;
            #pragma unroll
            for (int j = 0; j < 5; ++j) { e[j] = wmul[j]*expf(s[qi][j]-mx); den += e[j]; }
            const float inv = kp ? (1.f/den) : 0.f;
            #pragma unroll
            for (int d = 0; d < 4; ++d) {
                float o = 0.f;
                #pragma unroll
                for (int j = 0; j < 5; ++j) o += e[j]*v[j][d];
                AU[((size_t)a*2 + qi)*Hh + col0 + d] = o * inv;
            }
        }
    }
}

// ---------------- Phase 4: broadcast scatter to output ----------------
__global__ __launch_bounds__(256)
void scatter_kernel(const float* __restrict__ OU, float* __restrict__ out)
{
    const int a = blockIdx.x;
    const int b = a >> 9;
    const int n = a & 511;
    const float* r0 = OU + (size_t)(2*a)*Hh;
    const float* r1 = r0 + Hh;
    float* dst = out + ((size_t)b*(Nn*BS) + (size_t)n*BS)*Hh;
    for (int h = threadIdx.x; h < Hh; h += blockDim.x) {
        const float v0 = r0[h];
        const float v1 = r1[h];
        dst[h] = v0;
        #pragma unroll
        for (int j = 1; j < BS; ++j) dst[(size_t)j*Hh + h] = v1;
    }
}

// ---------------- launcher ----------------
extern "C" void kernel_launch(void* const* d_in, const int* in_sizes, int n_in,
                              void* d_out, int out_size, void* d_ws, size_t ws_size,
                              hipStream_t stream) {
    const int*           input_ids = (const int*)d_in[0];
    const float*         hidden    = (const float*)d_in[1];
    const int*           anchors   = (const int*)d_in[2];
    const unsigned char* keep      = (const unsigned char*)d_in[3];
    const float*         embed     = (const float*)d_in[4];
    const float*         Wq        = (const float*)d_in[5];
    const float*         Wk        = (const float*)d_in[6];
    const float*         Wv        = (const float*)d_in[7];
    const float*         Wo        = (const float*)d_in[8];
    float* out = (float*)d_out;

    float* ws = (float*)d_ws;
    const size_t uM = (size_t)NA * 5;         // 5120 unique rows
    float* U  = ws;
    float* Qb = U  + uM * Hh;
    float* Kb = Qb + uM * Hh;
    float* Vb = Kb + uM * Hh;
    float* AU = Vb + uM * Hh;                 // 2048 x 2048
    float* OU = AU + (size_t)NA*2 * Hh;       // 2048 x 2048

    build_u_kernel<<<NA, 256, 0, stream>>>(input_ids, hidden, anchors, keep, embed, U);

    dim3 gQKV(Hh/TN, (int)(uM/TM));           // (16, 80)
    gemm_f32_wmma_kernel<<<gQKV, 256, 0, stream>>>(U, Wq, Qb, (int)uM, Hh, Hh);
    gemm_f32_wmma_kernel<<<gQKV, 256, 0, stream>>>(U, Wk, Kb, (int)uM, Hh, Hh);
    gemm_f32_wmma_kernel<<<gQKV, 256, 0, stream>>>(U, Wv, Vb, (int)uM, Hh, Hh);

    attn_kernel<<<NA, 256, 0, stream>>>(Qb, Kb, Vb, keep, AU);

    dim3 gO(Hh/TN, (NA*2)/TM);                // (16, 32)
    gemm_f32_wmma_kernel<<<gO, 256, 0, stream>>>(AU, Wo, OU, NA*2, Hh, Hh);

    scatter_kernel<<<NA, 256, 0, stream>>>(OU, out);
}